// MultiLoraColumnParallelLinear_83623013253475
// MI455X (gfx1250) — compile-verified
//
#include <hip/hip_runtime.h>

// MultiLoRA column-parallel linear:  y[b,s,o] = sum_k x[b,s,k] * W[ids[b],o,k]
//   x: [4, 2048, 4096] f32   W: [32, 16, 4096] f32   ids: [4] i32
//   y: [4, 2048, 16] f32
//
// Memory-bound (AI ~= 8 flop/byte, ~136 MB traffic -> ~5.8us at 23.3 TB/s).
// fp32 WMMA 16x16x4 keeps exact reference precision; compute ceiling is far
// above what HBM can feed, so the kernel is organized purely around streaming
// x coalesced while B fragments (W) hit L2.

typedef float v2f __attribute__((ext_vector_type(2)));
typedef float v8f __attribute__((ext_vector_type(8)));

namespace {
constexpr int kDIn  = 4096;  // K
constexpr int kDOut = 16;    // N (= one WMMA tile)
constexpr int kSeq  = 2048;  // S
constexpr int kRowsPerWave = 32;  // two 16-row M-tiles per wave
}

__global__ __launch_bounds__(32)
void multilora_wmma_f32_kernel(const float* __restrict__ x,
                               const int*   __restrict__ adapter_ids,
                               const float* __restrict__ W,
                               float*       __restrict__ y) {
    const int lane = threadIdx.x & 31;
    const int half = lane >> 4;     // 0: lanes 0-15, 1: lanes 16-31
    const int r    = lane & 15;     // A: row-in-tile / B: output column / D: column

    // Absolute row range handled by this wave (x is contiguous [B*S, D_IN]).
    const int row0 = blockIdx.x * kRowsPerWave;
    const int b    = row0 / kSeq;           // batch index (uniform per block)
    const int aid  = adapter_ids[b];        // scalar load, block-uniform

    // A-fragment lane base: x[row0 + r][2*half + k]  (b64 per step)
    const float* xp0 = x + (size_t)(row0 + r) * kDIn + 2 * half;
    const float* xp1 = xp0 + (size_t)16 * kDIn;    // second 16-row tile
    // B-fragment lane base: W[aid][r][2*half + k]   (same b64 pattern, L2-hot)
    const float* wp  = W + ((size_t)aid * kDOut + r) * kDIn + 2 * half;

    v8f c0 = {};
    v8f c1 = {};

    // K loop: 1024 steps of K=4. Unroll x8 -> 24 outstanding b64 loads/wave.
    #pragma unroll 8
    for (int k = 0; k < kDIn; k += 4) {
        v2f bf = *(const v2f*)(wp  + k);
        v2f a0 = *(const v2f*)(xp0 + k);
        v2f a1 = *(const v2f*)(xp1 + k);
        // D = A(16x4) * B(4x16) + C ; 8 args:
        // (neg_a, A, neg_b, B, c_mod, C, reuse_a, reuse_b)
        c0 = __builtin_amdgcn_wmma_f32_16x16x4_f32(
                 false, a0, false, bf, (short)0, c0, false, false);
        c1 = __builtin_amdgcn_wmma_f32_16x16x4_f32(
                 false, a1, false, bf, (short)0, c1, false, false);
    }

    // D layout: VGPR v -> row (v + 8*half), col r. Store coalesced in o.
    float* yp0 = y + (size_t)(row0 + 8 * half) * kDOut + r;
    #pragma unroll
    for (int v = 0; v < 8; ++v) {
        yp0[(size_t)v * kDOut] = c0[v];
    }
    float* yp1 = yp0 + (size_t)16 * kDOut;   // second tile, rows +16
    #pragma unroll
    for (int v = 0; v < 8; ++v) {
        yp1[(size_t)v * kDOut] = c1[v];
    }
}

extern "C" void kernel_launch(void* const* d_in, const int* in_sizes, int n_in,
                              void* d_out, int out_size, void* d_ws, size_t ws_size,
                              hipStream_t stream) {
    (void)in_sizes; (void)n_in; (void)d_ws; (void)ws_size; (void)out_size;
    const float* x   = (const float*)d_in[0];   // [4, 2048, 4096] f32
    const int*   ids = (const int*)  d_in[1];   // [4] i32
    const float* W   = (const float*)d_in[2];   // [32, 16, 4096] f32
    float*       y   = (float*)d_out;           // [4, 2048, 16] f32

    constexpr int totalRows = 4 * 2048;
    dim3 grid(totalRows / kRowsPerWave);        // 256 single-wave workgroups
    dim3 block(32);                             // wave32
    multilora_wmma_f32_kernel<<<grid, block, 0, stream>>>(x, ids, W, y);
}